// MyEntmax15_43602507989425
// MI455X (gfx1250) — compile-verified
//
#include <hip/hip_runtime.h>
#include <math.h>

#define TPB   256              // 8 wave32s
#define COLS  4096
#define VPT   4                // float4 vectors per thread (4*256*4 = 4096 floats)
#define EPT   16               // elements per thread
#define RPB   4                // rows per block (pipelined)
#define BISECT_ITERS 16
#define NWAVES (TPB / 32)

#ifndef __has_builtin
#define __has_builtin(x) 0
#endif

#if __has_builtin(__builtin_amdgcn_global_load_async_to_lds_b128)
#define HAVE_ASYNC_LDS 1
#else
#define HAVE_ASYNC_LDS 0
#endif

#if HAVE_ASYNC_LDS
typedef __attribute__((ext_vector_type(4))) int v4i;
typedef __attribute__((address_space(1))) v4i gv4i;   // global int4
typedef __attribute__((address_space(3))) v4i lv4i;   // LDS int4
#endif

// ---------------------------------------------------------------------------
// Async staging: copy one 4096-float row from global into an LDS buffer.
// Each of the 256 threads moves 4 x b128 (its own 16 floats) via
// global_load_async_to_lds_b128 (ASYNCcnt) so the DMA overlaps the previous
// row's bisection compute. Fallback: plain ld/st through VGPRs.
// ---------------------------------------------------------------------------
__device__ __forceinline__ void async_copy_row(const float* __restrict__ src,
                                               float* lbuf, int tid)
{
#if HAVE_ASYNC_LDS
#pragma unroll
  for (int j = 0; j < VPT; ++j) {
    const int vec = j * TPB + tid;  // float4 index within the row
    __builtin_amdgcn_global_load_async_to_lds_b128(
        (gv4i*)(src + vec * 4),
        (lv4i*)(lbuf + vec * 4),
        0, 0);
  }
#else
#pragma unroll
  for (int j = 0; j < VPT; ++j) {
    const int vec = j * TPB + tid;
    ((float4*)lbuf)[vec] = ((const float4*)src)[vec];
  }
#endif
}

__device__ __forceinline__ void wait_async_copies()
{
#if HAVE_ASYNC_LDS
#if __has_builtin(__builtin_amdgcn_s_wait_asynccnt)
  __builtin_amdgcn_s_wait_asynccnt(0);
#else
  asm volatile("s_wait_asynccnt 0" ::: "memory");
#endif
#endif
}

// ---------------------------------------------------------------------------
// wave32 shuffles + single-barrier block reductions.
// Parity double-buffered partials: call i's reads are ordered before call
// (i+2)'s writes to the same buffer by call (i+1)'s barrier, so one
// __syncthreads per reduction suffices.
// ---------------------------------------------------------------------------
__device__ __forceinline__ float wave_sum(float v)
{
#pragma unroll
  for (int m = 16; m >= 1; m >>= 1) v += __shfl_xor(v, m, 32);
  return v;
}

__device__ __forceinline__ float wave_max(float v)
{
#pragma unroll
  for (int m = 16; m >= 1; m >>= 1) v = fmaxf(v, __shfl_xor(v, m, 32));
  return v;
}

__device__ __forceinline__ float block_sum(float v, float (*red)[NWAVES],
                                           int tid, int& pc)
{
  v = wave_sum(v);
  float* slot = red[pc & 1];
  if ((tid & 31) == 0) slot[tid >> 5] = v;
  __syncthreads();
  float t = slot[0];
#pragma unroll
  for (int w = 1; w < NWAVES; ++w) t += slot[w];
  ++pc;
  return t;
}

__device__ __forceinline__ float block_max(float v, float (*red)[NWAVES],
                                           int tid, int& pc)
{
  v = wave_max(v);
  float* slot = red[pc & 1];
  if ((tid & 31) == 0) slot[tid >> 5] = v;
  __syncthreads();
  float t = slot[0];
#pragma unroll
  for (int w = 1; w < NWAVES; ++w) t = fmaxf(t, slot[w]);
  ++pc;
  return t;
}

__device__ __forceinline__ void block_sum3(float& a, float& b, float& c,
                                           float (*red)[NWAVES][4],
                                           int tid, int& pc)
{
  a = wave_sum(a);
  b = wave_sum(b);
  c = wave_sum(c);
  float (*slot)[4] = red[pc & 1];
  if ((tid & 31) == 0) {
    const int w = tid >> 5;
    slot[w][0] = a; slot[w][1] = b; slot[w][2] = c;
  }
  __syncthreads();
  float ta = 0.0f, tb = 0.0f, tc = 0.0f;
#pragma unroll
  for (int w = 0; w < NWAVES; ++w) {
    ta += slot[w][0]; tb += slot[w][1]; tc += slot[w][2];
  }
  a = ta; b = tb; c = tc;
  ++pc;
}

// ---------------------------------------------------------------------------
// Exact entmax-1.5 per row, sort-free:
//   1) row max, z = (x - max) * 0.5                       (z_max == 0)
//   2) bisect f(tau) = sum (z-tau)_+^2 - 1 on [-1, 0]     (16 steps)
//   3) exact quadratic solve on support {z > tau_lo}:
//        tau* = mean - sqrt((1 - (q - k*mean^2)) / k)     (== reference formula)
//   4) out = (z - tau*)_+^2 / (norm + 1e-8),
//      norm = q - 2*tau*sz + k*tau^2  (analytic, from reduced stats)
// ---------------------------------------------------------------------------
__global__ __launch_bounds__(TPB) void entmax15_kernel(const float* __restrict__ x,
                                                       float* __restrict__ out,
                                                       int rows)
{
  __shared__ __align__(16) float buf[2][COLS];   // 32 KB double buffer
  __shared__ float redS[2][NWAVES];              // scalar reduction partials
  __shared__ float redV[2][NWAVES][4];           // fused 3-way partials

  const int tid  = threadIdx.x;
  const int row0 = blockIdx.x * RPB;
  if (row0 >= rows) return;

  int pc = 0;                                    // reduction parity counter

  async_copy_row(x + (size_t)row0 * COLS, buf[0], tid);

  for (int r = 0; r < RPB; ++r) {
    const int row = row0 + r;
    if (row >= rows) break;
    const int p = r & 1;

    wait_async_copies();     // my wave's b128 async transfers for buf[p] done
    __syncthreads();         // whole row resident in LDS

    if (r + 1 < RPB && row + 1 < rows)
      async_copy_row(x + (size_t)(row + 1) * COLS, buf[p ^ 1], tid);  // overlap

    // ---- pull my 16 elements into VGPRs, fold row max ----
    float z[EPT];
    float m = -__builtin_inff();
#pragma unroll
    for (int j = 0; j < VPT; ++j) {
      float4 v = ((const float4*)buf[p])[j * TPB + tid];
      z[j * 4 + 0] = v.x; z[j * 4 + 1] = v.y;
      z[j * 4 + 2] = v.z; z[j * 4 + 3] = v.w;
      m = fmaxf(m, fmaxf(fmaxf(v.x, v.y), fmaxf(v.z, v.w)));
    }
    m = block_max(m, redS, tid, pc);

#pragma unroll
    for (int e = 0; e < EPT; ++e) z[e] = (z[e] - m) * 0.5f;

    // ---- bisection: f(lo) >= 1 (max elem alone gives 1), f(hi=0) = 0 < 1 ----
    float lo = -1.0f, hi = 0.0f;
    for (int it = 0; it < BISECT_ITERS; ++it) {
      const float tau = 0.5f * (lo + hi);
      float s = 0.0f;
#pragma unroll
      for (int e = 0; e < EPT; ++e) {
        float d = fmaxf(z[e] - tau, 0.0f);
        s = fmaf(d, d, s);
      }
      s = block_sum(s, redS, tid, pc);     // broadcast -> uniform branch
      if (s >= 1.0f) lo = tau; else hi = tau;
    }

    // ---- exact quadratic solve on localized support {z > lo} ----
    float k = 0.0f, sz = 0.0f, sq = 0.0f;
#pragma unroll
    for (int e = 0; e < EPT; ++e) {
      const float a  = (z[e] > lo) ? 1.0f : 0.0f;
      const float az = a * z[e];
      k  += a;
      sz += az;
      sq  = fmaf(az, z[e], sq);
    }
    block_sum3(k, sz, sq, redV, tid, pc);

    const float mean  = sz / k;                       // k >= 1 (z_max=0 > lo)
    const float ss    = sq - k * mean * mean;         // k * running variance
    const float delta = (1.0f - ss) / k;
    const float tau_s = mean - sqrtf(fmaxf(delta, 0.0f));

    // analytic output norm: sum over support of (z - tau*)^2
    const float norm = fmaxf(fmaf(k, tau_s * tau_s, fmaf(-2.0f * tau_s, sz, sq)),
                             0.0f);
    const float inv  = 1.0f / (norm + 1e-8f);

    // ---- output ----
    float* __restrict__ orow = out + (size_t)row * COLS;
#pragma unroll
    for (int j = 0; j < VPT; ++j) {
      float4 v;
      float d0 = fmaxf(z[j * 4 + 0] - tau_s, 0.0f);
      float d1 = fmaxf(z[j * 4 + 1] - tau_s, 0.0f);
      float d2 = fmaxf(z[j * 4 + 2] - tau_s, 0.0f);
      float d3 = fmaxf(z[j * 4 + 3] - tau_s, 0.0f);
      v.x = d0 * d0 * inv; v.y = d1 * d1 * inv;
      v.z = d2 * d2 * inv; v.w = d3 * d3 * inv;
      ((float4*)orow)[j * TPB + tid] = v;
    }
  }
}

extern "C" void kernel_launch(void* const* d_in, const int* in_sizes, int n_in,
                              void* d_out, int out_size, void* d_ws, size_t ws_size,
                              hipStream_t stream)
{
  const float* x   = (const float*)d_in[0];
  float*       out = (float*)d_out;
  const int total  = in_sizes[0];
  const int rows   = total / COLS;           // 16384
  const int blocks = (rows + RPB - 1) / RPB; // 4096 blocks of 8 wave32s

  hipLaunchKernelGGL(entmax15_kernel, dim3(blocks), dim3(TPB), 0, stream,
                     x, out, rows);
}